// GCNII_11424613007593
// MI455X (gfx1250) — compile-verified
//
#include <hip/hip_runtime.h>
#include <math.h>

typedef __attribute__((ext_vector_type(2))) float v2f;
typedef __attribute__((ext_vector_type(8))) float v8f;

#define NODES   100000
#define HDIM    256
#define NLAYERS 8
#define KT      64          // K-chunk staged in LDS
#define LDSTRIDE 66         // padded (even -> keeps b64 alignment, breaks bank clash)

#if __has_builtin(__builtin_amdgcn_global_load_async_to_lds_b32) && \
    __has_builtin(__builtin_amdgcn_s_wait_asynccnt)
#define HAVE_ASYNC_LDS 1
typedef __attribute__((address_space(1))) int gas_int;   // global (AS1)
typedef __attribute__((address_space(3))) int las_int;   // LDS (AS3)
#else
#define HAVE_ASYNC_LDS 0
#endif

// deg[c] += 1 for every edge target (self-loop +1 folded into dinv kernel)
__global__ void degree_kernel(const int* __restrict__ cols, float* __restrict__ deg, int E) {
    int i = blockIdx.x * blockDim.x + threadIdx.x;
    if (i < E) atomicAdd(&deg[cols[i]], 1.0f);
}

__global__ void dinv_kernel(const float* __restrict__ deg, float* __restrict__ dinv, int n) {
    int i = blockIdx.x * blockDim.x + threadIdx.x;
    if (i < n) dinv[i] = rsqrtf(deg[i] + 1.0f);   // +1 for the self loop
}

// Edge-parallel SpMM: one wave per edge, lane covers 8 features (256/32).
// agg[c] += dinv[r]*dinv[c] * h[r]   (h rows are L2-resident: 102MB < 192MB L2)
__global__ void spmm_kernel(const int* __restrict__ rows, const int* __restrict__ cols,
                            const float* __restrict__ dinv, const float* __restrict__ Hin,
                            float* __restrict__ Agg, int E) {
    int lane   = threadIdx.x & 31;
    int gwave  = (blockIdx.x * blockDim.x + threadIdx.x) >> 5;
    int stride = (gridDim.x * blockDim.x) >> 5;
    for (int e = gwave; e < E; e += stride) {
        int r = rows[e];
        int c = cols[e];
        float w = dinv[r] * dinv[c];
        const float* hr = Hin + (size_t)r * HDIM;
        float*       ac = Agg + (size_t)c * HDIM;
#pragma unroll
        for (int i = 0; i < HDIM / 32; ++i) {
            int f = lane + i * 32;
            atomicAdd(ac + f, w * hr[f]);
        }
    }
}

// mid = (1-ALPHA)*(agg + dinv^2 * h) + ALPHA * h0      (in place on agg buffer)
__global__ void combine_kernel(float* __restrict__ Mid, const float* __restrict__ Hin,
                               const float* __restrict__ H0, const float* __restrict__ dinv,
                               int total) {
    int i = blockIdx.x * blockDim.x + threadIdx.x;
    if (i >= total) return;
    int node = i >> 8;                 // HDIM == 256
    float d = dinv[node];
    float v = Mid[i] + d * d * Hin[i];
    Mid[i] = 0.9f * v + 0.1f * H0[i];  // (1-ALPHA)=0.9, ALPHA=0.1
}

// Out = wAcc*(A@W) + wMid*Mid (+bias) (+relu).  A: MxK row-major, W: KxHDIM row-major.
// Block = 4 waves -> 64x64 output tile. Each wave: 16 rows x 64 cols (4 WMMA accs),
// so one A fragment feeds 4 v_wmma_f32_16x16x4_f32. W chunk staged in LDS (transposed).
__global__ void __launch_bounds__(128)
wmma_gemm_kernel(const float* __restrict__ A, const float* __restrict__ W,
                 const float* __restrict__ bias, const float* __restrict__ Mid,
                 float* __restrict__ Out, int M, int K,
                 float wMid, float wAcc, int do_relu) {
    __shared__ float ldsW[64 * LDSTRIDE];   // [c][k] transposed W chunk

    const int tid     = threadIdx.x;
    const int lane    = tid & 31;
    const int wave    = tid >> 5;
    const int rowBase = blockIdx.x * 64 + wave * 16;
    const int colBase = blockIdx.y * 64;
    const int l15     = lane & 15;
    const int khalf   = (lane >> 4) << 1;   // 0 (lanes 0-15) or 2 (lanes 16-31)

    int arow = rowBase + l15;
    if (arow >= M) arow = M - 1;            // clamp; only OOB output rows polluted
    const float* aPtr = A + (size_t)arow * K + khalf;

    // cooperative staging coords: 128 threads cover 64 cols x 2 k-rows per step
    const int cc = tid & 63;
    const int kb = tid >> 6;                // 0 or 1

    v8f acc[4] = {};

    for (int k0 = 0; k0 < K; k0 += KT) {
        __syncthreads();
        const float* wSrc = W + (size_t)(k0 + kb) * HDIM + colBase + cc;
#pragma unroll
        for (int i = 0; i < KT / 2; ++i) {
            int k = kb + 2 * i;
            // ldsW[cc][k] = W[k0+k][colBase+cc]
#if HAVE_ASYNC_LDS
            __builtin_amdgcn_global_load_async_to_lds_b32(
                (gas_int*)(wSrc + (size_t)2 * i * HDIM),
                (las_int*)&ldsW[cc * LDSTRIDE + k],
                0, 0);
#else
            ldsW[cc * LDSTRIDE + k] = wSrc[(size_t)2 * i * HDIM];
#endif
        }
#if HAVE_ASYNC_LDS
        __builtin_amdgcn_s_wait_asynccnt(0);
#endif
        __syncthreads();

        if (k0 + KT < K) __builtin_prefetch(aPtr + k0 + KT, 0, 1);  // next A chunk

#pragma unroll 4
        for (int k = 0; k < KT; k += 4) {
            v2f a = *(const v2f*)(aPtr + k0 + k);     // K = khalf+k, khalf+k+1
            const int kk = k + khalf;
#pragma unroll
            for (int j = 0; j < 4; ++j) {
                v2f b = *(const v2f*)&ldsW[(j * 16 + l15) * LDSTRIDE + kk];
                acc[j] = __builtin_amdgcn_wmma_f32_16x16x4_f32(
                    false, a, false, b, (short)0, acc[j], false, false);
            }
        }
    }

    const int row0 = rowBase + ((lane >> 4) << 3);    // +0 or +8
#pragma unroll
    for (int j = 0; j < 4; ++j) {
        const int col = colBase + j * 16 + l15;
        const float bb = bias ? bias[col] : 0.0f;
#pragma unroll
        for (int v = 0; v < 8; ++v) {
            int row = row0 + v;
            if (row < M) {
                float o = wAcc * acc[j][v] + bb;
                if (Mid) o += wMid * Mid[(size_t)row * HDIM + col];
                if (do_relu) o = fmaxf(o, 0.0f);
                Out[(size_t)row * HDIM + col] = o;
            }
        }
    }
}

extern "C" void kernel_launch(void* const* d_in, const int* in_sizes, int n_in,
                              void* d_out, int out_size, void* d_ws, size_t ws_size,
                              hipStream_t stream) {
    const float* x       = (const float*)d_in[0];
    const int*   ei      = (const int*)d_in[1];
    const float* W_embed = (const float*)d_in[2];
    const float* b_embed = (const float*)d_in[3];
    const float* W_convs = (const float*)d_in[4];

    const int    Nn = NODES;
    const int    E  = in_sizes[1] / 2;
    const int    IN = in_sizes[0] / Nn;          // 512
    const size_t NH = (size_t)Nn * HDIM;

    const int* rows = ei;          // edge_index[0] = source
    const int* cols = ei + E;      // edge_index[1] = target

    float* ws   = (float*)d_ws;
    float* deg  = ws;
    float* dinv = ws + Nn;
    float* h0   = ws + 2 * (size_t)Nn;
    float* hA   = h0 + NH;
    float* hB   = hA + NH;
    float* mid  = hB + NH;

    (void)hipMemsetAsync(deg, 0, (size_t)Nn * sizeof(float), stream);
    degree_kernel<<<(E + 255) / 256, 256, 0, stream>>>(cols, deg, E);
    dinv_kernel<<<(Nn + 255) / 256, 256, 0, stream>>>(deg, dinv, Nn);

    dim3 ggrid((Nn + 63) / 64, HDIM / 64);
    // h0 = relu(x @ W_embed + b)
    wmma_gemm_kernel<<<ggrid, 128, 0, stream>>>(x, W_embed, b_embed, nullptr, h0,
                                                Nn, IN, 0.0f, 1.0f, 1);

    const float* hcur = h0;
    float* pp[2] = {hA, hB};
    for (int l = 1; l <= NLAYERS; ++l) {
        (void)hipMemsetAsync(mid, 0, NH * sizeof(float), stream);
        spmm_kernel<<<2048, 256, 0, stream>>>(rows, cols, dinv, hcur, mid, E);
        combine_kernel<<<(int)((NH + 255) / 256), 256, 0, stream>>>(mid, hcur, h0, dinv, (int)NH);
        float beta = logf(0.5f / (float)l + 1.0f);   // log(THETA/l + 1)
        float* outp = (l == NLAYERS) ? (float*)d_out : pp[(l - 1) & 1];
        wmma_gemm_kernel<<<ggrid, 128, 0, stream>>>(mid, W_convs + (size_t)(l - 1) * HDIM * HDIM,
                                                    nullptr, mid, outp, Nn, HDIM,
                                                    1.0f - beta, beta, (l < NLAYERS) ? 1 : 0);
        hcur = outp;
    }
    (void)n_in; (void)out_size; (void)ws_size;
}